// GCN_reg_64278480552405
// MI455X (gfx1250) — compile-verified
//
#include <hip/hip_runtime.h>

typedef __attribute__((ext_vector_type(2))) float v2f;
typedef __attribute__((ext_vector_type(8))) float v8f;

// ---------------------------------------------------------------------------
// CSR build utilities
// ---------------------------------------------------------------------------
__global__ void k_zero_i32(int* __restrict__ p, int n) {
    int i = blockIdx.x * blockDim.x + threadIdx.x;
    if (i < n) p[i] = 0;
}

__global__ void k_count_deg(const int* __restrict__ dst, int E, int* __restrict__ cnt) {
    int e = blockIdx.x * blockDim.x + threadIdx.x;
    if (e < E) atomicAdd(&cnt[dst[e]], 1);
}

// Single-workgroup chunked Hillis-Steele exclusive scan over cnt[0..n).
__global__ void k_scan(const int* __restrict__ cnt, int* __restrict__ rowstart,
                       int* __restrict__ cursor, float* __restrict__ dinv, int n) {
    __shared__ int buf[1024];
    __shared__ int s_carry;
    int t = threadIdx.x;
    if (t == 0) s_carry = 0;
    __syncthreads();
    for (int base = 0; base < n; base += 1024) {
        int idx = base + t;
        int v = (idx < n) ? cnt[idx] : 0;
        buf[t] = v;
        __syncthreads();
        for (int off = 1; off < 1024; off <<= 1) {
            int x = (t >= off) ? buf[t - off] : 0;
            __syncthreads();
            buf[t] += x;
            __syncthreads();
        }
        int carry = s_carry;
        int incl  = buf[t];
        if (idx < n) {
            int excl = carry + incl - v;
            rowstart[idx] = excl;
            cursor[idx]   = excl;
            dinv[idx]     = rsqrtf((float)(v + 1));   // +1 self loop; always > 0
        }
        __syncthreads();
        if (t == 1023) s_carry = carry + incl;
        __syncthreads();
    }
    if (t == 0) rowstart[n] = s_carry;
}

__global__ void k_fill_csr(const int* __restrict__ src, const int* __restrict__ dst, int E,
                           int* __restrict__ cursor, const float* __restrict__ dinv,
                           int* __restrict__ csr_src, float* __restrict__ csr_w) {
    int e = blockIdx.x * blockDim.x + threadIdx.x;
    if (e >= E) return;
    int s = src[e];
    int d = dst[e];
    int pos = atomicAdd(&cursor[d], 1);
    csr_src[pos] = s;
    csr_w[pos]   = dinv[s] * dinv[d];
}

// ---------------------------------------------------------------------------
// Repack B[K][N] (row major) into fragment-native layout:
//   Bp[(k/4)*N*4 + n*4 + (k%4)]
// so a WMMA B-fragment (lane-half h needs rows k+2h, k+2h+1 at one column)
// becomes a single aligned b64 load.
// ---------------------------------------------------------------------------
__global__ void k_pack_b(const float* __restrict__ B, float* __restrict__ Bp, int N, int K) {
    int idx = blockIdx.x * blockDim.x + threadIdx.x;
    if (idx >= K * N) return;
    int k = idx / N;
    int n = idx - k * N;
    Bp[((size_t)(k >> 2) * N + n) * 4 + (k & 3)] = B[idx];
}

// ---------------------------------------------------------------------------
// fp32 WMMA GEMM: C[M,N] = A[M,K] @ B[K,N], B pre-packed as above.
// One wave per 16x64 output tile; K stepped by 4 with V_WMMA_F32_16X16X4_F32.
// Distance-2 software pipeline: fragment sets for k+4 and k+8 are kept in
// flight, each WMMA group consumes loads issued two iterations earlier, so
// s_wait_loadcnt sits a full loop body away from its loads.
// Layouts per CDNA5 ISA 7.12.2 (lane-half h, vgpr v):
//   A: K = v + 2h, M = lane%16   B: K = v + 2h, N = lane%16   C/D: M = v + 8h
// ---------------------------------------------------------------------------
__global__ void k_gemm_wmma(const float* __restrict__ A, const float* __restrict__ Bp,
                            float* __restrict__ C, int M, int N, int K) {
    const int lane = threadIdx.x & 31;
    const int half = lane >> 4;    // 0 or 1
    const int r    = lane & 15;
    const int tm   = blockIdx.y * 16;
    const int tn   = blockIdx.x * 64;

    int arow_idx = tm + r;
    if (arow_idx >= M) arow_idx = M - 1;                 // safe clamp (M%16==0 here)
    const float* arow    = A + (size_t)arow_idx * K + 2 * half;
    const float* bp_base = Bp + (size_t)(tn + r) * 4 + 2 * half;
    const size_t kgrp_stride = (size_t)N * 4;            // floats per k-group of 4

    v8f c0 = {}, c1 = {}, c2 = {}, c3 = {};

    // prologue: fragment sets for k = 0 and k = 4 (K >= 8 guaranteed: K % 64 == 0)
    v2f a0  = *(const v2f*)(arow);
    v2f b00 = *(const v2f*)(bp_base);
    v2f b01 = *(const v2f*)(bp_base + 64);
    v2f b02 = *(const v2f*)(bp_base + 128);
    v2f b03 = *(const v2f*)(bp_base + 192);

    const float* bp1 = bp_base + kgrp_stride;
    v2f a1  = *(const v2f*)(arow + 4);
    v2f b10 = *(const v2f*)(bp1);
    v2f b11 = *(const v2f*)(bp1 + 64);
    v2f b12 = *(const v2f*)(bp1 + 128);
    v2f b13 = *(const v2f*)(bp1 + 192);

#pragma unroll 2
    for (int k = 8; k < K; k += 4) {
        // issue loads for step k (2 steps ahead of what we consume now)
        const float* bp = bp_base + (size_t)(k >> 2) * kgrp_stride;
        v2f a2  = *(const v2f*)(arow + k);
        v2f b20 = *(const v2f*)(bp);
        v2f b21 = *(const v2f*)(bp + 64);
        v2f b22 = *(const v2f*)(bp + 128);
        v2f b23 = *(const v2f*)(bp + 192);

        // consume fragments loaded two iterations ago
        c0 = __builtin_amdgcn_wmma_f32_16x16x4_f32(false, a0, false, b00, (short)0, c0, false, false);
        c1 = __builtin_amdgcn_wmma_f32_16x16x4_f32(false, a0, false, b01, (short)0, c1, false, false);
        c2 = __builtin_amdgcn_wmma_f32_16x16x4_f32(false, a0, false, b02, (short)0, c2, false, false);
        c3 = __builtin_amdgcn_wmma_f32_16x16x4_f32(false, a0, false, b03, (short)0, c3, false, false);

        // rotate pipeline registers
        a0 = a1;  b00 = b10; b01 = b11; b02 = b12; b03 = b13;
        a1 = a2;  b10 = b20; b11 = b21; b12 = b22; b13 = b23;
    }
    // epilogue: two steps still in flight
    c0 = __builtin_amdgcn_wmma_f32_16x16x4_f32(false, a0, false, b00, (short)0, c0, false, false);
    c1 = __builtin_amdgcn_wmma_f32_16x16x4_f32(false, a0, false, b01, (short)0, c1, false, false);
    c2 = __builtin_amdgcn_wmma_f32_16x16x4_f32(false, a0, false, b02, (short)0, c2, false, false);
    c3 = __builtin_amdgcn_wmma_f32_16x16x4_f32(false, a0, false, b03, (short)0, c3, false, false);

    c0 = __builtin_amdgcn_wmma_f32_16x16x4_f32(false, a1, false, b10, (short)0, c0, false, false);
    c1 = __builtin_amdgcn_wmma_f32_16x16x4_f32(false, a1, false, b11, (short)0, c1, false, false);
    c2 = __builtin_amdgcn_wmma_f32_16x16x4_f32(false, a1, false, b12, (short)0, c2, false, false);
    c3 = __builtin_amdgcn_wmma_f32_16x16x4_f32(false, a1, false, b13, (short)0, c3, false, false);

#pragma unroll
    for (int v = 0; v < 8; ++v) {
        int m = tm + v + 8 * half;
        if (m < M) {
            float* crow = C + (size_t)m * N + tn + r;
            crow[0]  = c0[v];
            crow[16] = c1[v];
            crow[32] = c2[v];
            crow[48] = c3[v];
        }
    }
}

// ---------------------------------------------------------------------------
// Gather-based GCN aggregation (float4 vectorized):
//   out[n,f] = sum_{e: dst=n} h[src_e,f]*norm_e + h[n,f]*dinv[n]^2 + bias[f]
// One workgroup (128 threads) per node; each thread owns one float4 chunk,
// accumulator lives in registers across the (uniform) edge loop.
// ---------------------------------------------------------------------------
__global__ void k_aggregate(const float* __restrict__ h, const int* __restrict__ csr_src,
                            const float* __restrict__ csr_w, const int* __restrict__ rowstart,
                            const float* __restrict__ dinv, const float* __restrict__ bias,
                            float* __restrict__ out, int H, int do_relu) {
    const int n   = blockIdx.x;
    const float d = dinv[n];
    const float selfw = d * d;
    const int beg = rowstart[n];
    const int end = rowstart[n + 1];
    const int H4  = H >> 2;

    const float4* h4 = (const float4*)h;
    const float4* b4 = (const float4*)bias;
    float4*       o4 = (float4*)out;

    for (int f = threadIdx.x; f < H4; f += blockDim.x) {
        float4 v = h4[(size_t)n * H4 + f];
        float4 acc;
        acc.x = v.x * selfw; acc.y = v.y * selfw; acc.z = v.z * selfw; acc.w = v.w * selfw;
        for (int i = beg; i < end; ++i) {
            int s   = csr_src[i];
            float w = csr_w[i];
            float4 u = h4[(size_t)s * H4 + f];
            acc.x += u.x * w; acc.y += u.y * w; acc.z += u.z * w; acc.w += u.w * w;
        }
        float4 bb = b4[f];
        acc.x += bb.x; acc.y += bb.y; acc.z += bb.z; acc.w += bb.w;
        if (do_relu) {
            acc.x = fmaxf(acc.x, 0.0f); acc.y = fmaxf(acc.y, 0.0f);
            acc.z = fmaxf(acc.z, 0.0f); acc.w = fmaxf(acc.w, 0.0f);
        }
        o4[(size_t)n * H4 + f] = acc;
    }
}

// ---------------------------------------------------------------------------
// Fused global_add_pool + linear head (float4 vectorized). batch is sorted:
// binary-search each graph's node range; no atomics -> deterministic.
// ---------------------------------------------------------------------------
__global__ void k_pool_head(const float* __restrict__ a2, const int* __restrict__ batch,
                            const float* __restrict__ Wlin, const float* __restrict__ blin,
                            float* __restrict__ out, int n_nodes, int H) {
    const int g = blockIdx.x;

    int lo = 0, hi = n_nodes;
    while (lo < hi) { int mid = (lo + hi) >> 1; if (batch[mid] < g) lo = mid + 1; else hi = mid; }
    const int start = lo;
    hi = n_nodes;
    while (lo < hi) { int mid = (lo + hi) >> 1; if (batch[mid] < g + 1) lo = mid + 1; else hi = mid; }
    const int end = lo;

    const int H4 = H >> 2;
    const float4* a4 = (const float4*)a2;
    const float4* w4 = (const float4*)Wlin;

    float partial = 0.0f;
    for (int f = threadIdx.x; f < H4; f += blockDim.x) {
        float4 acc = make_float4(0.f, 0.f, 0.f, 0.f);
        for (int n = start; n < end; ++n) {
            float4 u = a4[(size_t)n * H4 + f];
            acc.x += u.x; acc.y += u.y; acc.z += u.z; acc.w += u.w;
        }
        float4 wl = w4[f];
        partial += acc.x * wl.x + acc.y * wl.y + acc.z * wl.z + acc.w * wl.w;
    }

    __shared__ float red[128];
    red[threadIdx.x] = partial;
    __syncthreads();
    for (int s = 64; s > 0; s >>= 1) {
        if (threadIdx.x < s) red[threadIdx.x] += red[threadIdx.x + s];
        __syncthreads();
    }
    if (threadIdx.x == 0) out[g] = red[0] + blin[0];
}

// ---------------------------------------------------------------------------
extern "C" void kernel_launch(void* const* d_in, const int* in_sizes, int n_in,
                              void* d_out, int out_size, void* d_ws, size_t ws_size,
                              hipStream_t stream) {
    const float* x    = (const float*)d_in[0];
    const int*   ei   = (const int*)d_in[1];
    const int*   bat  = (const int*)d_in[2];
    // d_in[3] = num_graphs (device scalar); num_graphs == out_size on host side
    const float* W1   = (const float*)d_in[4];
    const float* b1   = (const float*)d_in[5];
    const float* W2   = (const float*)d_in[6];
    const float* b2   = (const float*)d_in[7];
    const float* Wlin = (const float*)d_in[8];
    const float* blin = (const float*)d_in[9];
    float* out        = (float*)d_out;

    const int n_nodes = in_sizes[2];           // 10000
    const int E       = in_sizes[1] / 2;       // 160000
    const int H       = in_sizes[5];           // 512 (hidden)
    const int D       = in_sizes[0] / n_nodes; // 512 (d_in)
    const int G       = out_size;              // num_graphs = 64

    const int* src = ei;
    const int* dst = ei + E;

    // ---- workspace carve (256B aligned) ----
    char* ws = (char*)d_ws;
    auto carve = [&](size_t bytes) {
        char* p = ws;
        ws += (bytes + 255) & ~(size_t)255;
        return p;
    };
    float* hbuf     = (float*)carve((size_t)n_nodes * H * sizeof(float)); // GEMM outputs
    float* abuf     = (float*)carve((size_t)n_nodes * H * sizeof(float)); // aggregated outputs
    float* Bp       = (float*)carve((size_t)H * H * sizeof(float));       // packed weights
    int*   cnt      = (int*)  carve((size_t)n_nodes * sizeof(int));
    int*   rowstart = (int*)  carve((size_t)(n_nodes + 1) * sizeof(int));
    int*   cursor   = (int*)  carve((size_t)n_nodes * sizeof(int));
    float* dinv     = (float*)carve((size_t)n_nodes * sizeof(float));
    int*   csr_src  = (int*)  carve((size_t)E * sizeof(int));
    float* csr_w    = (float*)carve((size_t)E * sizeof(float));
    (void)ws_size;

    // ---- build normalization + CSR (dst-grouped) ----
    k_zero_i32<<<(n_nodes + 255) / 256, 256, 0, stream>>>(cnt, n_nodes);
    k_count_deg<<<(E + 255) / 256, 256, 0, stream>>>(dst, E, cnt);
    k_scan<<<1, 1024, 0, stream>>>(cnt, rowstart, cursor, dinv, n_nodes);
    k_fill_csr<<<(E + 255) / 256, 256, 0, stream>>>(src, dst, E, cursor, dinv, csr_src, csr_w);

    dim3 ggrid(H / 64, (n_nodes + 15) / 16);

    // ---- layer 1: h = x @ W1 ; a1 = relu(agg(h) + b1) ----
    k_pack_b<<<(D * H + 255) / 256, 256, 0, stream>>>(W1, Bp, H, D);
    k_gemm_wmma<<<ggrid, 32, 0, stream>>>(x, Bp, hbuf, n_nodes, H, D);
    k_aggregate<<<n_nodes, 128, 0, stream>>>(hbuf, csr_src, csr_w, rowstart, dinv, b1, abuf, H, 1);

    // ---- layer 2: h = a1 @ W2 ; a2 = agg(h) + b2 ----
    k_pack_b<<<(H * H + 255) / 256, 256, 0, stream>>>(W2, Bp, H, H);
    k_gemm_wmma<<<ggrid, 32, 0, stream>>>(abuf, Bp, hbuf, n_nodes, H, H);
    k_aggregate<<<n_nodes, 128, 0, stream>>>(hbuf, csr_src, csr_w, rowstart, dinv, b2, abuf, H, 0);

    // ---- pooled @ Wlin + blin ----
    k_pool_head<<<G, 128, 0, stream>>>(abuf, bat, Wlin, blin, out, n_nodes, H);
}